// ProtoLayer_5789615915304
// MI455X (gfx1250) — compile-verified
//
#include <hip/hip_runtime.h>

typedef float v2f  __attribute__((ext_vector_type(2)));
typedef float v8f  __attribute__((ext_vector_type(8)));
typedef float f32x4 __attribute__((ext_vector_type(4)));

#define T_      16
#define WAY_    20
#define SHOT_   5
#define QUERY_  30
#define C_      640
#define WQ_     (WAY_ * QUERY_)   // 600
#define WPAD_   32                // way padded to 2 WMMA N tiles
#define MTILE_  16
#define KCHUNK_ 64
#define LDSROW_ (KCHUNK_ + 2)     // 66: breaks the 256B bank period (64 banks x 4B)
#define KSPLIT_ 2
#define KHALF_  (C_ / KSPLIT_)    // 320 channels per wave
#define PAIRS_  4                 // M tiles per block
#define NWAVES_ (PAIRS_ * KSPLIT_)// 8 waves per block
#define MTILES_ ((WQ_ + MTILE_ - 1) / MTILE_)   // 38

// ---------------------------------------------------------------------------
// Kernel 1: proto[t][wpad][c] = mean over shot (zero for wpad >= 20),
//           pp[t][wpad] = ||proto||^2.  One wave32 per (t, wpad) row.
// ---------------------------------------------------------------------------
__global__ __launch_bounds__(32)
void proto_kernel(const float* __restrict__ support,
                  float* __restrict__ proto, float* __restrict__ pp) {
    const int b    = blockIdx.x;          // 0 .. T_*WPAD_-1
    const int t    = b >> 5;
    const int w    = b & 31;
    const int lane = threadIdx.x;         // 0..31

    float sq = 0.f;
    float* prow = proto + ((size_t)t * WPAD_ + w) * C_;
    if (w < WAY_) {
        const float* sp = support + ((size_t)t * WAY_ * SHOT_ + (size_t)w * SHOT_) * C_;
        for (int col = lane; col < C_; col += 32) {
            float acc = 0.f;
#pragma unroll
            for (int s = 0; s < SHOT_; ++s) acc += sp[s * C_ + col];
            acc *= (1.0f / SHOT_);
            prow[col] = acc;
            sq += acc * acc;
        }
    } else {
        for (int col = lane; col < C_; col += 32) prow[col] = 0.f;
    }
#pragma unroll
    for (int off = 16; off > 0; off >>= 1) sq += __shfl_xor(sq, off, 32);
    if (lane == 0) pp[t * WPAD_ + w] = sq;
}

// ---------------------------------------------------------------------------
// Kernel 2: out[t][m][n] = 2*(q . proto) - ||q||^2 - ||p||^2
// Two waves cooperate per 16-row M tile (split-K over the 640 channels) to
// double wave-level parallelism for HBM latency hiding; partials combined
// through LDS.  Matrix core: V_WMMA_F32_16X16X4_F32.
// ---------------------------------------------------------------------------
__global__ __launch_bounds__(32 * NWAVES_)
void dist_kernel(const float* __restrict__ q, const float* __restrict__ proto,
                 const float* __restrict__ pp, float* __restrict__ out) {
    __shared__ float stage[NWAVES_][MTILE_ * LDSROW_];   // 8 x 4.125 KB A staging
    __shared__ float cacc[PAIRS_][32][17];               // kh=1 partial D (padded)
    __shared__ float cqq[PAIRS_][KSPLIT_][MTILE_];       // per-row ||q||^2 halves

    const int lane  = threadIdx.x;        // 0..31
    const int wv    = threadIdx.y;        // 0..7
    const int pair  = wv >> 1;            // which M tile in block
    const int kh    = wv & 1;             // which K half
    const int t     = blockIdx.y;
    const int mtile = blockIdx.x * PAIRS_ + pair;
    const int m0    = mtile * MTILE_;

    const int row   = lane & 15;          // M for A-frag / N for B-frag
    const int khalf = lane >> 4;          // 0: K=k,k+1   1: K=k+2,k+3

    const float* qbase = q     + (size_t)t * WQ_   * C_;
    const float* pbase = proto + (size_t)t * WPAD_ * C_;

    v8f acc0 = {0.f, 0.f, 0.f, 0.f, 0.f, 0.f, 0.f, 0.f};
    v8f acc1 = acc0;
    float qqp = 0.f;

    const int kbeg = kh * KHALF_;
    for (int k0 = kbeg; k0 < kbeg + KHALF_; k0 += KCHUNK_) {
        // Stage A chunk (16 rows x 64 K) into LDS with coalesced b128 loads,
        // zero-filling rows >= 600 so EXEC stays all-1s through the WMMAs.
#pragma unroll
        for (int i = 0; i < 8; ++i) {
            int idx = i * 32 + lane;            // 0..255 quads
            int r   = idx >> 4;                 // 0..15
            int k4  = (idx & 15) << 2;          // 0..60 step 4
            int gm  = m0 + r;
            f32x4 v = {0.f, 0.f, 0.f, 0.f};
            if (gm < WQ_)
                v = *(const f32x4*)(qbase + (size_t)gm * C_ + k0 + k4);
            *(f32x4*)(&stage[wv][r * LDSROW_ + k4]) = v;
        }
        __syncthreads();

#pragma unroll
        for (int ks = 0; ks < KCHUNK_; ks += 4) {
            // A fragment (16x4 f32): 2 consecutive K per lane, conflict-free
            // thanks to the 66-float row stride.
            v2f a = *(const v2f*)(&stage[wv][row * LDSROW_ + ks + khalf * 2]);
            // B fragments (4x16 f32): B[k][n] = proto[n][k]; row-major proto
            // gives each lane its 2 consecutive K values in one b64 load
            // (L2-resident: proto is 80 KB per t).
            const int kg = k0 + ks + khalf * 2;
            v2f b0 = *(const v2f*)(pbase + (size_t)row        * C_ + kg);
            v2f b1 = *(const v2f*)(pbase + (size_t)(row + 16) * C_ + kg);

            // ||q||^2 partial: lanes l and l+16 cover disjoint K of row l&15.
            qqp += a.x * a.x + a.y * a.y;

            acc0 = __builtin_amdgcn_wmma_f32_16x16x4_f32(
                false, a, false, b0, (short)0, acc0, false, false);
            acc1 = __builtin_amdgcn_wmma_f32_16x16x4_f32(
                false, a, false, b1, (short)0, acc1, false, false);
        }
        __syncthreads();
    }

    // Per-row ||q||^2 for this K half (combine lane halves), publish to LDS.
    float qqh = qqp + __shfl_xor(qqp, 16, 32);
    if (lane < 16) cqq[pair][kh][lane] = qqh;
    // kh=1 wave exports its partial accumulators for the pair's kh=0 wave.
    if (kh == 1) {
#pragma unroll
        for (int r = 0; r < 8; ++r) {
            cacc[pair][lane][r]     = acc0[r];
            cacc[pair][lane][8 + r] = acc1[r];
        }
    }
    __syncthreads();

    if (kh == 0) {
#pragma unroll
        for (int r = 0; r < 8; ++r) {
            acc0[r] += cacc[pair][lane][r];
            acc1[r] += cacc[pair][lane][8 + r];
        }
        const int   n0  = lane & 15;
        const float pp0 = pp[t * WPAD_ + n0];
        const float pp1 = pp[t * WPAD_ + 16 + n0];
        float* obase = out + (size_t)t * WQ_ * WAY_;

#pragma unroll
        for (int r = 0; r < 8; ++r) {
            const int   ml  = r + khalf * 8;   // local M of acc element r
            const int   gm  = m0 + ml;
            const float qqm = cqq[pair][0][ml] + cqq[pair][1][ml];
            if (gm < WQ_) {
                obase[(size_t)gm * WAY_ + n0] = 2.f * acc0[r] - qqm - pp0;
                const int n1 = 16 + n0;
                if (n1 < WAY_)
                    obase[(size_t)gm * WAY_ + n1] = 2.f * acc1[r] - qqm - pp1;
            }
        }
    }
}

// ---------------------------------------------------------------------------
extern "C" void kernel_launch(void* const* d_in, const int* in_sizes, int n_in,
                              void* d_out, int out_size, void* d_ws, size_t ws_size,
                              hipStream_t stream) {
    const float* qf = (const float*)d_in[0];   // (16, 600, 640) f32
    const float* sf = (const float*)d_in[1];   // (16, 100, 640) f32
    // d_in[2..4] = way/shot/query scalars; hardcoded as compile-time constants.

    float* proto = (float*)d_ws;                       // T_*WPAD_*C_ floats (1.25 MB)
    float* pp    = proto + (size_t)T_ * WPAD_ * C_;    // T_*WPAD_ floats

    proto_kernel<<<dim3(T_ * WPAD_), dim3(32), 0, stream>>>(sf, proto, pp);

    dim3 grid((MTILES_ + PAIRS_ - 1) / PAIRS_, T_);    // (10, 16) -> 1280 waves
    dist_kernel<<<grid, dim3(32, NWAVES_), 0, stream>>>(qf, proto, pp, (float*)d_out);
}